// GraphSAGE_34626026341043
// MI455X (gfx1250) — compile-verified
//
#include <hip/hip_runtime.h>
#include <hip/hip_bf16.h>
#include <stdint.h>

// ---------------------------------------------------------------------------
// GraphSAGE SAGEConv for MI455X (gfx1250, wave32)
//   out = (mean_{j in N(i)} x_j) @ W_l^T + b_l + x_i @ W_r^T
// N=100000 nodes, E=1600000 edges, C_in=32, C_out=128, all fp32.
//
// Pipeline (all on `stream`):
//   k0: zero summed[N,32] + cnt[N] in d_ws
//   k1: scatter - 4 edges per wave, coalesced NT index loads + shfl broadcast,
//       lane==channel, per-channel global_atomic_add_f32 (L2 resident)
//   k2: fused mean + dual-GEMM via V_WMMA_F32_16X16X4_F32:
//       A = [summed*rinv | x] (16x64 tile in LDS), B = [W_l^T ; W_r^T],
//       C preloaded with b_l; NT stores for the write-once output.
// ---------------------------------------------------------------------------

typedef __attribute__((ext_vector_type(2))) float v2f;
typedef __attribute__((ext_vector_type(8))) float v8f;

#define IN_CH   32
#define OUT_CH  128
#define K_TOT   64          // concatenated K = 2 * IN_CH
#define LDS_STRIDE 66       // 16x64 A tile padded: bank = (2m+k)%64 -> conflict-free

__global__ void sage_zero_kernel(float* __restrict__ p, int n) {
    int i = blockIdx.x * blockDim.x + threadIdx.x;
    if (i < n) p[i] = 0.0f;
}

// 4 edges per wave; lanes 0-3 load src ids, lanes 4-7 load dst ids (coalesced,
// non-temporal: edge list is streamed exactly once), then shfl-broadcast.
// lane == channel for the gather + atomic scatter-add.
__global__ void sage_scatter_kernel(const float* __restrict__ x,
                                    const long long* __restrict__ edge_index,
                                    float* __restrict__ summed,
                                    float* __restrict__ cnt,
                                    int nEdges) {
    const int lane   = threadIdx.x & 31;
    const int waveId = blockIdx.x * (blockDim.x >> 5) + (threadIdx.x >> 5);
    const int e0     = waveId << 2;              // first of 4 edges
    if (e0 >= nEdges) return;

    long long idx64 = 0;
    if (lane < 4) {
        int e = e0 + lane;
        if (e < nEdges) idx64 = __builtin_nontemporal_load(&edge_index[e]);           // src
    } else if (lane < 8) {
        int e = e0 + lane - 4;
        if (e < nEdges) idx64 = __builtin_nontemporal_load(&edge_index[(size_t)nEdges + e]); // dst
    }
    const int idx = (int)idx64;   // node ids < 100000 fit in 32 bits

    #pragma unroll
    for (int i = 0; i < 4; ++i) {
        const int e = e0 + i;                    // uniform per wave
        if (e >= nEdges) break;
        const int src = __shfl(idx, i, 32);
        const int dst = __shfl(idx, 4 + i, 32);
        const float v = x[src * IN_CH + lane];   // one 128B cacheline per edge
        atomicAdd(&summed[dst * IN_CH + lane], v);
        if (lane == 0) atomicAdd(&cnt[dst], 1.0f);
    }
}

// 128 threads = 4 waves. Block handles 16 node-rows x all 128 out-channels.
// Mean-division fused into the A-tile load; each wave computes two 16x16
// output tiles via 16 chained WMMA_F32_16X16X4_F32.
__global__ __launch_bounds__(128) void sage_gemm_kernel(
        const float* __restrict__ summed, // [N,32] scatter-summed neighbors
        const float* __restrict__ cnt,    // [N]    neighbor counts (float)
        const float* __restrict__ x,      // [N,32] root features
        const float* __restrict__ Wl,     // [128,32]
        const float* __restrict__ Wr,     // [128,32]
        const float* __restrict__ bl,     // [128]
        float* __restrict__ out,          // [N,128]
        int nNodes) {
    __shared__ float At[16 * LDS_STRIDE];
    __shared__ float rinv[16];

    const int tid = threadIdx.x;
    const int block_row = blockIdx.x * 16;

    // Per-row 1/max(cnt,1) for the mean (16 divides instead of 512)
    if (tid < 16) {
        int row = block_row + tid;
        float c = (row < nNodes) ? cnt[row] : 1.0f;
        rinv[tid] = 1.0f / fmaxf(c, 1.0f);
    }
    __syncthreads();

    // Cooperative load of A tile: rows = 16 nodes, cols = [mean(0..31) | x(32..63)]
    #pragma unroll
    for (int j = 0; j < 8; ++j) {
        int i = tid + 128 * j;          // 0..1023
        int r = i >> 6;                 // node row in tile
        int c = i & 63;                 // concatenated K column
        int row = block_row + r;
        float v = 0.0f;
        if (row < nNodes)
            v = (c < IN_CH) ? summed[row * IN_CH + c] * rinv[r]
                            : x[row * IN_CH + (c - IN_CH)];
        At[r * LDS_STRIDE + c] = v;
    }
    __syncthreads();

    const int wave = tid >> 5;
    const int lane = tid & 31;
    const int m    = lane & 15;              // A row for this lane
    const int koff = (lane >> 4) << 1;       // 0 for lanes 0-15, 2 for lanes 16-31
    const int mhi  = (lane >> 4) << 3;       // C/D: high lanes hold M+8
    const bool full_tile = (block_row + 16 <= nNodes);   // block-uniform

    #pragma unroll
    for (int t = 0; t < 2; ++t) {
        const int nt = wave + 4 * t;         // 8 N-tiles across 4 waves x 2
        const int n  = nt * 16 + (lane & 15);

        // C init: bias broadcast over M
        v8f c;
        const float bias = bl[n];
        #pragma unroll
        for (int r = 0; r < 8; ++r) c[r] = bias;

        // K loop over concatenated [W_l^T ; W_r^T]; a K-group of 4 never
        // straddles the 32 boundary, so pick the matrix once per step.
        #pragma unroll
        for (int kb = 0; kb < K_TOT; kb += 4) {
            const int k0 = kb + koff;
            v2f a;
            a.x = At[m * LDS_STRIDE + k0];
            a.y = At[m * LDS_STRIDE + k0 + 1];

            const float* W = (k0 < IN_CH) ? Wl : Wr;
            const int kk = k0 & (IN_CH - 1);
            v2f b;
            b.x = W[n * IN_CH + kk];
            b.y = W[n * IN_CH + kk + 1];

            // D = A(16x4) x B(4x16) + C, fp32 exact
            c = __builtin_amdgcn_wmma_f32_16x16x4_f32(
                    /*neg_a=*/false, a, /*neg_b=*/false, b,
                    /*c_mod=*/(short)0, c, /*reuse_a=*/false, /*reuse_b=*/false);
        }

        // Store D: lane -> N, vgpr r -> M (+8 for high half-wave).
        // Uniform fast path: no per-row exec juggling; NT stores (write-once).
        if (full_tile) {
            #pragma unroll
            for (int r = 0; r < 8; ++r)
                __builtin_nontemporal_store(c[r], &out[(block_row + r + mhi) * OUT_CH + n]);
        } else {
            #pragma unroll
            for (int r = 0; r < 8; ++r) {
                int row = block_row + r + mhi;
                if (row < nNodes) out[row * OUT_CH + n] = c[r];
            }
        }
    }
}

extern "C" void kernel_launch(void* const* d_in, const int* in_sizes, int n_in,
                              void* d_out, int out_size, void* d_ws, size_t ws_size,
                              hipStream_t stream) {
    const float*     x  = (const float*)d_in[0];
    const long long* ei = (const long long*)d_in[1];   // int64 [2, E]
    const float*     Wl = (const float*)d_in[2];
    const float*     Wr = (const float*)d_in[3];
    const float*     bl = (const float*)d_in[4];
    float* out = (float*)d_out;

    const int nNodes = in_sizes[0] / IN_CH;
    const int nEdges = in_sizes[1] / 2;

    float* summed = (float*)d_ws;                       // [nNodes*32]
    float* cnt    = summed + (size_t)nNodes * IN_CH;    // [nNodes]

    // k0: zero accumulators
    {
        int n = nNodes * (IN_CH + 1);
        sage_zero_kernel<<<(n + 255) / 256, 256, 0, stream>>>(summed, n);
    }
    // k1: edge scatter (8 waves/block * 4 edges/wave = 32 edges per block)
    {
        int blocks = (nEdges + 31) / 32;
        sage_scatter_kernel<<<blocks, 256, 0, stream>>>(x, ei, summed, cnt, nEdges);
    }
    // k2: fused mean + dual GEMM + bias via WMMA
    {
        int blocks = (nNodes + 15) / 16;
        sage_gemm_kernel<<<blocks, 128, 0, stream>>>(summed, cnt, x, Wl, Wr, bl, out, nNodes);
    }
}